// Model_292057776820
// MI455X (gfx1250) — compile-verified
//
#include <hip/hip_runtime.h>
#include <hip/hip_bf16.h>

// ---------------- problem constants (match reference) ----------------
constexpr int kN    = 50000;
constexpr int kE    = 800000;
constexpr int kEP   = kE + kN;     // edges + self loops
constexpr int kWID  = 128;         // H*C, also F_IN
constexpr int kH    = 4;
constexpr int kB    = 128;
constexpr int kNHID = 256;
constexpr int kNOUT = 768;
constexpr float kBNScale = 0.99999500003749969f;  // 1/sqrt(1+1e-5)
constexpr float kNegSlope = 0.2f;

typedef float v2f __attribute__((ext_vector_type(2)));
typedef float v8f __attribute__((ext_vector_type(8)));
typedef int vint4 __attribute__((vector_size(16)));  // matches builtin param type

#if __has_builtin(__builtin_amdgcn_global_load_async_to_lds_b128) && \
    __has_builtin(__builtin_amdgcn_s_wait_asynccnt)
#define HAVE_ASYNC_LDS 1
#else
#define HAVE_ASYNC_LDS 0
#endif

// ---------------- WMMA 16x16 tile GEMM core (global A, global B) -----------
// D(16x16) += A(16xK) * B(Kx16); A row-major lda, B row-major ldb.
// A 16x4 f32 frag: lane m=lane&15, koff=(lane>>4)*2, a={A[m][koff],A[m][koff+1]}.
// B frag mirrors A transposed (col n=lane&15). C/D: VGPR r -> row r+(lane>=16)*8.
__device__ inline v8f wmma_tile(const float* __restrict__ A, int lda,
                                const float* __restrict__ Bm, int ldb,
                                int K, int lane) {
  v8f c = {};
  const int m = lane & 15;
  const int koff = (lane >> 4) * 2;
  for (int k = 0; k < K; k += 4) {
    v2f a, b;
    a[0] = A[m * lda + k + koff];
    a[1] = A[m * lda + k + koff + 1];
    b[0] = Bm[(k + koff) * ldb + m];
    b[1] = Bm[(k + koff + 1) * ldb + m];
    c = __builtin_amdgcn_wmma_f32_16x16x4_f32(false, a, false, b,
                                              (short)0, c, false, false);
  }
  return c;
}

// ---------------- kernels ----------------

// XW[N,128] = Hin[N,128] @ W[128,128].
// Block = 256 threads (8 waves); W (64KB) staged in LDS once per block via
// gfx1250 async global->LDS loads; each wave computes one 16x16 col tile,
// B fragments come from LDS (ds path), A fragments stream from global.
__global__ void k_gemm_node(const float* __restrict__ A,
                            const float* __restrict__ W,
                            float* __restrict__ C) {
  __shared__ float sW[kWID * kWID];  // 64 KB (of 320 KB/WGP)

#if HAVE_ASYNC_LDS
  {
    __attribute__((address_space(1))) vint4* g =
        (__attribute__((address_space(1))) vint4*)W;
    __attribute__((address_space(3))) vint4* l =
        (__attribute__((address_space(3))) vint4*)sW;
    const int t = threadIdx.x;
#pragma unroll
    for (int i = 0; i < 16; ++i) {
      const int idx = i * 256 + t;  // 16B units
      __builtin_amdgcn_global_load_async_to_lds_b128(g + idx, l + idx, 0, 0);
    }
    __builtin_amdgcn_s_wait_asynccnt(0);
  }
#else
  for (int i = threadIdx.x; i < (kWID * kWID) / 4; i += 256) {
    ((float4*)sW)[i] = ((const float4*)W)[i];
  }
#endif
  __syncthreads();

  const int lane = threadIdx.x & 31;
  const int wv   = threadIdx.x >> 5;   // col tile 0..7
  const int rowt = blockIdx.x;         // row tile
  const int m    = lane & 15;
  const int koff = (lane >> 4) * 2;
  const float* Ap = A + (size_t)rowt * 16 * kWID + (size_t)m * kWID;
  const float* Bs = sW + wv * 16 + m;

  v8f acc = {};
#pragma unroll 8
  for (int k = 0; k < kWID; k += 4) {
    v2f a, b;
    a[0] = Ap[k + koff];
    a[1] = Ap[k + koff + 1];
    b[0] = Bs[(k + koff) * kWID];
    b[1] = Bs[(k + koff + 1) * kWID];
    acc = __builtin_amdgcn_wmma_f32_16x16x4_f32(false, a, false, b,
                                                (short)0, acc, false, false);
  }

  const int col = wv * 16 + m;
  const int rb  = (lane >> 4) * 8;
#pragma unroll
  for (int r = 0; r < 8; ++r) {
    C[(size_t)(rowt * 16 + rb + r) * kWID + col] = acc[r];
  }
}

// generic small GEMM with bias (+optional relu): C[M,Ntot] = A[M,K] @ B[K,Ntot] + bias
__global__ void k_gemm_fc(const float* __restrict__ A,
                          const float* __restrict__ Bm,
                          const float* __restrict__ bias,
                          float* __restrict__ C,
                          int K, int Ntot, int relu) {
  const int lane = threadIdx.x & 31;   // block = 32 (one wave)
  const int mt = blockIdx.x, nt = blockIdx.y;
  v8f acc = wmma_tile(A + (size_t)mt * 16 * K, K, Bm + nt * 16, Ntot, K, lane);
  const int col = nt * 16 + (lane & 15);
  const int rb  = (lane >> 4) * 8;
#pragma unroll
  for (int r = 0; r < 8; ++r) {
    float v = acc[r] + bias[col];
    if (relu) v = fmaxf(v, 0.f);
    C[(size_t)(mt * 16 + rb + r) * Ntot + col] = v;
  }
}

// per-node attention scores: ssrc[n,h]=sum_c XW[n,h*32+c]*as[h*32+c], same for ad
__global__ void k_scores(const float* __restrict__ XW,
                         const float* __restrict__ as_,
                         const float* __restrict__ ad_,
                         float* __restrict__ ssrc,
                         float* __restrict__ sdst) {
  int n = blockIdx.x * blockDim.x + threadIdx.x;
  if (n >= kN) return;
  const float* row = XW + (size_t)n * kWID;
#pragma unroll
  for (int h = 0; h < kH; ++h) {
    float s = 0.f, d = 0.f;
    for (int c = 0; c < 32; ++c) {
      float v = row[h * 32 + c];
      s += v * as_[h * 32 + c];
      d += v * ad_[h * 32 + c];
    }
    ssrc[n * kH + h] = s;
    sdst[n * kH + h] = d;
  }
}

__global__ void k_init_layer(float* __restrict__ num,
                             float* __restrict__ mmax,
                             float* __restrict__ den) {
  int i = blockIdx.x * blockDim.x + threadIdx.x;
  if (i < kN * kWID) num[i] = 0.f;
  if (i < kN * kH) { mmax[i] = -1e30f; den[i] = 0.f; }
}

__device__ inline float lrelu(float x) { return x > 0.f ? x : kNegSlope * x; }

__device__ inline void atomicMaxF(float* a, float v) {
  if (v >= 0.f) atomicMax((int*)a, __float_as_int(v));
  else          atomicMin((unsigned int*)a, __float_as_uint(v));
}

// pass 1: segment max over dst per head
__global__ void k_edge_max(const int* __restrict__ ei,
                           const float* __restrict__ ssrc,
                           const float* __restrict__ sdst,
                           float* __restrict__ mmax) {
  int e = blockIdx.x * blockDim.x + threadIdx.x;
  if (e >= kEP) return;
  int s = (e < kE) ? ei[e] : (e - kE);
  int d = (e < kE) ? ei[kE + e] : (e - kE);
#pragma unroll
  for (int h = 0; h < kH; ++h) {
    float v = lrelu(ssrc[s * kH + h] + sdst[d * kH + h]);
    atomicMaxF(&mmax[d * kH + h], v);
  }
}

// pass 2: one wave per edge; lane handles 4 channels. num += w*xs[src]; den += w.
__global__ void k_edge_acc(const int* __restrict__ ei,
                           const float* __restrict__ XW,
                           const float* __restrict__ ssrc,
                           const float* __restrict__ sdst,
                           const float* __restrict__ mmax,
                           float* __restrict__ den,
                           float* __restrict__ num) {
  long long t = (long long)blockIdx.x * blockDim.x + threadIdx.x;
  long long e = t >> 5;
  int lane = (int)(t & 31);
  if (e >= kEP) return;
  int s = (e < kE) ? ei[e] : (int)(e - kE);
  int d = (e < kE) ? ei[kE + e] : (int)(e - kE);
  int h = lane >> 3;                      // 8 lanes per head
  float sc = lrelu(ssrc[s * kH + h] + sdst[d * kH + h]);
  float w  = __expf(sc - mmax[d * kH + h]);
  if ((lane & 7) == 0) atomicAdd(&den[d * kH + h], w);
  int c0 = lane * 4;
  const float4 xs = *(const float4*)(XW + (size_t)s * kWID + c0);
  float* np = num + (size_t)d * kWID + c0;
  atomicAdd(np + 0, w * xs.x);
  atomicAdd(np + 1, w * xs.y);
  atomicAdd(np + 2, w * xs.z);
  atomicAdd(np + 3, w * xs.w);
}

// out = relu(((num/den)+gat_bias) * g*BN_SCALE + be), in place on num
__global__ void k_finalize(float* __restrict__ num,
                           const float* __restrict__ den,
                           const float* __restrict__ bias,
                           const float* __restrict__ g,
                           const float* __restrict__ be) {
  int i = blockIdx.x * blockDim.x + threadIdx.x;
  if (i >= kN * kWID) return;
  int n = i >> 7, c = i & 127;
  float v = num[i] / (den[n * kH + (c >> 5)] + 1e-16f) + bias[c];
  v = v * (g[c] * kBNScale) + be[c];
  num[i] = fmaxf(v, 0.f);
}

__global__ void k_pool_init(float* __restrict__ pool, float* __restrict__ cnt) {
  int i = blockIdx.x * blockDim.x + threadIdx.x;
  if (i < kB * kWID) pool[i] = 0.f;
  if (i < kB) cnt[i] = 0.f;
}

__global__ void k_pool_acc(const float* __restrict__ h,
                           const int* __restrict__ batch,
                           float* __restrict__ pool,
                           float* __restrict__ cnt) {
  int i = blockIdx.x * blockDim.x + threadIdx.x;
  if (i >= kN * kWID) return;
  int n = i >> 7, c = i & 127;
  int b = batch[n];
  atomicAdd(&pool[b * kWID + c], h[i]);
  if (c == 0) atomicAdd(&cnt[b], 1.0f);
}

__global__ void k_pool_div(float* __restrict__ pool, const float* __restrict__ cnt) {
  int i = blockIdx.x * blockDim.x + threadIdx.x;
  if (i >= kB * kWID) return;
  pool[i] /= fmaxf(cnt[i >> 7], 1.f);
}

// ---------------- launcher ----------------
extern "C" void kernel_launch(void* const* d_in, const int* in_sizes, int n_in,
                              void* d_out, int out_size, void* d_ws, size_t ws_size,
                              hipStream_t stream) {
  const float* x     = (const float*)d_in[0];
  const int*   ei    = (const int*)d_in[1];
  const int*   batch = (const int*)d_in[2];
  const float* Wl[3]  = {(const float*)d_in[3], (const float*)d_in[9],  (const float*)d_in[15]};
  const float* asl[3] = {(const float*)d_in[4], (const float*)d_in[10], (const float*)d_in[16]};
  const float* adl[3] = {(const float*)d_in[5], (const float*)d_in[11], (const float*)d_in[17]};
  const float* bl[3]  = {(const float*)d_in[6], (const float*)d_in[12], (const float*)d_in[18]};
  const float* gl[3]  = {(const float*)d_in[7], (const float*)d_in[13], (const float*)d_in[19]};
  const float* bel[3] = {(const float*)d_in[8], (const float*)d_in[14], (const float*)d_in[20]};
  const float* fcW1 = (const float*)d_in[21];
  const float* fcb1 = (const float*)d_in[22];
  const float* fcW2 = (const float*)d_in[23];
  const float* fcb2 = (const float*)d_in[24];
  float* out = (float*)d_out;

  // workspace carve (floats)
  float* ws   = (float*)d_ws;
  float* XW   = ws;                         // N*128 (per-layer GEMM scratch)
  float* hb   = XW  + (size_t)kN * kWID;    // N*128 ping
  float* hc   = hb  + (size_t)kN * kWID;    // N*128 pong
  float* ssrc = hc  + (size_t)kN * kWID;    // N*4
  float* sdst = ssrc + (size_t)kN * kH;     // N*4
  float* mmax = sdst + (size_t)kN * kH;     // N*4
  float* den  = mmax + (size_t)kN * kH;     // N*4
  float* pool = den  + (size_t)kN * kH;     // B*128
  float* cnt  = pool + (size_t)kB * kWID;   // B
  float* fc1  = cnt  + kB;                  // B*256

  const int nThreads = 256;
  const int gN    = (kN + nThreads - 1) / nThreads;
  const int gNW   = (kN * kWID + nThreads - 1) / nThreads;
  const int gEP   = (kEP + nThreads - 1) / nThreads;
  const int gEP32 = (int)(((long long)kEP * 32 + nThreads - 1) / nThreads);
  const int gBW   = (kB * kWID + nThreads - 1) / nThreads;
  const int rowTiles = kN / 16;  // 3125 exactly

  const float* hin = x;
  float* houts[3] = {hb, hc, hb};
  for (int l = 0; l < 3; ++l) {
    float* hout = houts[l];
    k_gemm_node<<<rowTiles, 256, 0, stream>>>(hin, Wl[l], XW);
    k_scores<<<gN, nThreads, 0, stream>>>(XW, asl[l], adl[l], ssrc, sdst);
    k_init_layer<<<gNW, nThreads, 0, stream>>>(hout, mmax, den);
    k_edge_max<<<gEP, nThreads, 0, stream>>>(ei, ssrc, sdst, mmax);
    k_edge_acc<<<gEP32, nThreads, 0, stream>>>(ei, XW, ssrc, sdst, mmax, den, hout);
    k_finalize<<<gNW, nThreads, 0, stream>>>(hout, den, bl[l], gl[l], bel[l]);
    hin = hout;
  }

  k_pool_init<<<gBW, nThreads, 0, stream>>>(pool, cnt);
  k_pool_acc<<<gNW, nThreads, 0, stream>>>(hin, batch, pool, cnt);
  k_pool_div<<<gBW, nThreads, 0, stream>>>(pool, cnt);

  // FC1: [128,128]@[128,256]+b, relu ; FC2: [128,256]@[256,768]+b
  k_gemm_fc<<<dim3(kB / 16, kNHID / 16), 32, 0, stream>>>(pool, fcW1, fcb1, fc1, kWID, kNHID, 1);
  k_gemm_fc<<<dim3(kB / 16, kNOUT / 16), 32, 0, stream>>>(fc1, fcW2, fcb2, out, kNHID, kNOUT, 0);
}